// UKF_37357625541262
// MI455X (gfx1250) — compile-verified
//
#include <hip/hip_runtime.h>
#include <stdint.h>

// ---------------------------------------------------------------------------
// UKF, batch=16384, n=4, m=2, s=100.  Memory-bound (~160 MB total traffic ->
// ~7us at 23.3 TB/s).  One wave32 lane per filter; TDM (tensor_load_to_lds)
// double-buffers the strided measurement tensor into LDS per-wave; all output
// stores are 16B-aligned b128 with non-temporal hints (write-once streams,
// keep the 192MB L2 for the input streams instead).
// ---------------------------------------------------------------------------

#ifndef __has_builtin
#define __has_builtin(x) 0
#endif

#if __has_builtin(__builtin_amdgcn_tensor_load_to_lds)
#define UKF_TDM 1
#endif
#if __has_builtin(__builtin_amdgcn_s_wait_tensorcnt)
#define UKF_TCNT 1
#endif

typedef unsigned int ukf_u32x4 __attribute__((ext_vector_type(4)));
typedef int          ukf_i32x4 __attribute__((ext_vector_type(4)));
typedef int          ukf_i32x8 __attribute__((ext_vector_type(8)));
typedef float        f32x4    __attribute__((ext_vector_type(4)));

#define DT_C  0.1f
#define W0_C  0.2f        // LAM/(n+LAM) = 1/5
#define WI_C  0.1f        // 0.5/(n+LAM)
#define GAM_C 2.2360680f  // fp32(sqrt(5))

__device__ __forceinline__ void motion4(const float s[4], float hx, float hy,
                                        float uxdt, float uydt, float o[4]) {
  o[0] = s[0] + s[2] * DT_C + hx;
  o[1] = s[1] + s[3] * DT_C + hy;
  o[2] = s[2] + uxdt;
  o[3] = s[3] + uydt;
}

__device__ __forceinline__ f32x4 mkf4(const float a[4]) {
  f32x4 v; v[0] = a[0]; v[1] = a[1]; v[2] = a[2]; v[3] = a[3]; return v;
}

// streaming (non-temporal) 128-bit IO
__device__ __forceinline__ void st_nt(float* p, f32x4 v) {
  __builtin_nontemporal_store(v, reinterpret_cast<f32x4*>(p));
}
__device__ __forceinline__ f32x4 ld_nt(const float* p) {
  return __builtin_nontemporal_load(reinterpret_cast<const f32x4*>(p));
}

#ifdef UKF_TDM
// 2D TDM load: tile (tiled0 x tiled1) of 4-byte elements, row stride `stride0`
// elements, global tile start `g`, into LDS byte offset `lds_off`.
// D# packing per CDNA5 ISA ch.8 (group0: count/lds/global/type, group1: dims).
__device__ __forceinline__ void tdm_load_2d(unsigned lds_off, const float* g,
                                            int tiled0, int tiled1, int stride0) {
  unsigned long long ga = (unsigned long long)(uintptr_t)g;
  ukf_u32x4 g0;
  g0[0] = 1u;                                             // count=1, user D#
  g0[1] = lds_off;                                        // LDS byte address
  g0[2] = (unsigned)ga;                                   // global_addr[31:0]
  g0[3] = (unsigned)((ga >> 32) & 0x01FFFFFFull) | (2u << 30); // [56:32]|type=2
  ukf_i32x8 g1;
  g1[0] = (int)(2u << 16);          // workgroup_mask=0, data_size=2 (4 bytes)
  g1[1] = tiled0 << 16;             // tensor_dim0[15:0]  (== tile: no OOB)
  g1[2] = tiled1 << 16;             // tensor_dim1[15:0]
  g1[3] = tiled0 << 16;             // tile_dim0
  g1[4] = tiled1;                   // tile_dim1 (tile_dim2=0)
  g1[5] = stride0;                  // tensor_dim0_stride[31:0]
  g1[6] = 0;
  g1[7] = 0;
  ukf_i32x4 z4 = {0, 0, 0, 0};
#if __has_include(<hip/amd_detail/amd_gfx1250_TDM.h>)
  ukf_i32x8 z8 = {0, 0, 0, 0, 0, 0, 0, 0};
  __builtin_amdgcn_tensor_load_to_lds(g0, g1, z4, z4, z8, 0); // clang-23 6-arg
#else
  __builtin_amdgcn_tensor_load_to_lds(g0, g1, z4, z4, 0);     // ROCm7.2 5-arg
#endif
}
#endif

extern "C" __global__ __launch_bounds__(256, 1)
void ukf_cdna5_kernel(const float* __restrict__ meas,    // (B,2,S)
                      const float* __restrict__ state0,  // (B,4)
                      const float* __restrict__ cov0,    // (B,4,4)
                      const float* __restrict__ ctrl,    // (B,S,2)
                      const float* __restrict__ Qin,     // (4,4)
                      const float* __restrict__ Rin,     // (2,2)
                      float* __restrict__ outPred,       // (B,2,S)
                      float* __restrict__ outState,      // (B,4,S)
                      float* __restrict__ outCov,        // (B,4,4,S)
                      int B, int S) {
#ifdef UKF_TDM
  __shared__ float smem[2 * 8 * 256];  // [buf][wave][64 rows * 4 cols] = 16 KB
#endif
  const int tid  = (int)threadIdx.x;
  const int wave = tid >> 5;
  const int lane = tid & 31;
  const int b    = (int)blockIdx.x * 256 + tid;
  const int nchunks = S >> 2;
  if (b >= B) return;
  (void)wave; (void)lane;

  // Q (symmetric: 10 uniques) and R
  const float q00 = Qin[0],  q01 = Qin[1],  q02 = Qin[2],  q03 = Qin[3];
  const float q11 = Qin[5],  q12 = Qin[6],  q13 = Qin[7];
  const float q22 = Qin[10], q23 = Qin[11], q33 = Qin[15];
  const float r00 = Rin[0],  r01 = Rin[1],  r11 = Rin[3];

  float x[4];
  {
    f32x4 xv = ld_nt(state0 + (size_t)b * 4);
    x[0] = xv[0]; x[1] = xv[1]; x[2] = xv[2]; x[3] = xv[3];
  }
  float P[4][4];
#pragma unroll
  for (int k = 0; k < 4; ++k) {
    f32x4 pv = ld_nt(cov0 + (size_t)b * 16 + k * 4);
    P[k][0] = pv[0]; P[k][1] = pv[1]; P[k][2] = pv[2]; P[k][3] = pv[3];
  }

  const float* ctrlB = ctrl + (size_t)b * (size_t)(2 * S);

#ifdef UKF_TDM
  // Per-wave measurement tile: 64 rows (32 lanes x {zx,zy}) x 4 steps.
  const float* measWave =
      meas + ((size_t)blockIdx.x * 256 + (size_t)wave * 32) * 2u * (size_t)S;
  float* buf0 = &smem[(0 * 8 + wave) * 256];
  float* buf1 = &smem[(1 * 8 + wave) * 256];
  const unsigned off0 = (unsigned)(uintptr_t)buf0;  // low 32 bits = LDS offset
  const unsigned off1 = (unsigned)(uintptr_t)buf1;
  tdm_load_2d(off0, measWave, 4, 64, S);
#ifdef UKF_TCNT
  __builtin_amdgcn_s_wait_tensorcnt(0);
#endif
  asm volatile("" ::: "memory");
#endif

#pragma unroll 1
  for (int c = 0; c < nchunks; ++c) {
    float zx[4], zy[4];
#ifdef UKF_TDM
    if (c + 1 < nchunks)  // kick off next tile while we compute on this one
      tdm_load_2d(((c + 1) & 1) ? off1 : off0, measWave + (c + 1) * 4, 4, 64, S);
    {
      const float* buf = (c & 1) ? buf1 : buf0;
      f32x4 zxv = *reinterpret_cast<const f32x4*>(buf + lane * 8);
      f32x4 zyv = *reinterpret_cast<const f32x4*>(buf + lane * 8 + 4);
      zx[0] = zxv[0]; zx[1] = zxv[1]; zx[2] = zxv[2]; zx[3] = zxv[3];
      zy[0] = zyv[0]; zy[1] = zyv[1]; zy[2] = zyv[2]; zy[3] = zyv[3];
    }
#else
    {
      f32x4 zxv = ld_nt(meas + (size_t)b * 2 * S + c * 4);
      f32x4 zyv = ld_nt(meas + (size_t)b * 2 * S + S + c * 4);
      zx[0] = zxv[0]; zx[1] = zxv[1]; zx[2] = zxv[2]; zx[3] = zxv[3];
      zy[0] = zyv[0]; zy[1] = zyv[1]; zy[2] = zyv[2]; zy[3] = zyv[3];
    }
#endif
    f32x4 u01 = ld_nt(ctrlB + c * 8);
    f32x4 u23 = ld_nt(ctrlB + c * 8 + 4);
    if (c + 1 < nchunks) __builtin_prefetch(ctrlB + (c + 1) * 8, 0, 1);
    const float uu[4][2] = {{u01[0], u01[1]}, {u01[2], u01[3]},
                            {u23[0], u23[1]}, {u23[2], u23[3]}};

    float prd[2][4], st[4][4], cv[10][4];

#pragma unroll
    for (int j = 0; j < 4; ++j) {
      const float ux = uu[j][0], uy = uu[j][1];
      const float hx = 0.5f * ux * DT_C * DT_C;
      const float hy = 0.5f * uy * DT_C * DT_C;
      const float uxdt = ux * DT_C, uydt = uy * DT_C;

      // ---- Cholesky of P (SPD 4x4) ----
      float l00 = sqrtf(P[0][0]);            float i0 = 1.0f / l00;
      float l10 = P[1][0] * i0, l20 = P[2][0] * i0, l30 = P[3][0] * i0;
      float l11 = sqrtf(P[1][1] - l10 * l10); float i1 = 1.0f / l11;
      float l21 = (P[2][1] - l20 * l10) * i1;
      float l31 = (P[3][1] - l30 * l10) * i1;
      float l22 = sqrtf(P[2][2] - l20 * l20 - l21 * l21); float i2 = 1.0f / l22;
      float l32 = (P[3][2] - l30 * l20 - l31 * l21) * i2;
      float l33 = sqrtf(P[3][3] - l30 * l30 - l31 * l31 - l32 * l32);

      const float gof[4][4] = {  // gof[i][k] = gamma * L[k][i]
          {GAM_C * l00, GAM_C * l10, GAM_C * l20, GAM_C * l30},
          {0.0f,        GAM_C * l11, GAM_C * l21, GAM_C * l31},
          {0.0f,        0.0f,        GAM_C * l22, GAM_C * l32},
          {0.0f,        0.0f,        0.0f,        GAM_C * l33}};

      // ---- pass A: predicted mean over 9 sigma points ----
      float Fc[4]; motion4(x, hx, hy, uxdt, uydt, Fc);
      float xp[4];
#pragma unroll
      for (int k = 0; k < 4; ++k) xp[k] = W0_C * Fc[k];
#pragma unroll
      for (int i = 0; i < 4; ++i) {
        float sp[4], sm[4], Fp[4], Fm[4];
#pragma unroll
        for (int k = 0; k < 4; ++k) { sp[k] = x[k] + gof[i][k]; sm[k] = x[k] - gof[i][k]; }
        motion4(sp, hx, hy, uxdt, uydt, Fp);
        motion4(sm, hx, hy, uxdt, uydt, Fm);
#pragma unroll
        for (int k = 0; k < 4; ++k) xp[k] += WI_C * (Fp[k] + Fm[k]);
      }

      // ---- pass B: M = sum Wc dX dX^T (recompute sigma pts; regs stay low).
      //      P_pred = M+Q, S = M[:2,:2]+R, Pxz = M[:,:2]  (exact identities).
      float d0 = Fc[0] - xp[0], d1 = Fc[1] - xp[1], d2 = Fc[2] - xp[2], d3 = Fc[3] - xp[3];
      float M00 = W0_C * d0 * d0, M01 = W0_C * d0 * d1, M02 = W0_C * d0 * d2, M03 = W0_C * d0 * d3;
      float M11 = W0_C * d1 * d1, M12 = W0_C * d1 * d2, M13 = W0_C * d1 * d3;
      float M22 = W0_C * d2 * d2, M23 = W0_C * d2 * d3, M33 = W0_C * d3 * d3;
#pragma unroll
      for (int i = 0; i < 4; ++i) {
        float sp[4], sm[4], Fp[4], Fm[4];
#pragma unroll
        for (int k = 0; k < 4; ++k) { sp[k] = x[k] + gof[i][k]; sm[k] = x[k] - gof[i][k]; }
        motion4(sp, hx, hy, uxdt, uydt, Fp);
        motion4(sm, hx, hy, uxdt, uydt, Fm);
        float a0 = Fp[0] - xp[0], a1 = Fp[1] - xp[1], a2 = Fp[2] - xp[2], a3 = Fp[3] - xp[3];
        float b0 = Fm[0] - xp[0], b1 = Fm[1] - xp[1], b2 = Fm[2] - xp[2], b3 = Fm[3] - xp[3];
        M00 += WI_C * (a0 * a0 + b0 * b0); M01 += WI_C * (a0 * a1 + b0 * b1);
        M02 += WI_C * (a0 * a2 + b0 * b2); M03 += WI_C * (a0 * a3 + b0 * b3);
        M11 += WI_C * (a1 * a1 + b1 * b1); M12 += WI_C * (a1 * a2 + b1 * b2);
        M13 += WI_C * (a1 * a3 + b1 * b3); M22 += WI_C * (a2 * a2 + b2 * b2);
        M23 += WI_C * (a2 * a3 + b2 * b3); M33 += WI_C * (a3 * a3 + b3 * b3);
      }

      // ---- gain K = Pxz S^-1 (2x2 solve) ----
      float S00 = M00 + r00, S01 = M01 + r01, S11 = M11 + r11;
      float idet = 1.0f / (S00 * S11 - S01 * S01);
      float K00 = (M00 * S11 - M01 * S01) * idet, K01 = (M01 * S00 - M00 * S01) * idet;
      float K10 = (M01 * S11 - M11 * S01) * idet, K11 = (M11 * S00 - M01 * S01) * idet;
      float K20 = (M02 * S11 - M12 * S01) * idet, K21 = (M12 * S00 - M02 * S01) * idet;
      float K30 = (M03 * S11 - M13 * S01) * idet, K31 = (M13 * S00 - M03 * S01) * idet;

      float in0 = zx[j] - xp[0], in1 = zy[j] - xp[1];
      x[0] = xp[0] + K00 * in0 + K01 * in1;
      x[1] = xp[1] + K10 * in0 + K11 * in1;
      x[2] = xp[2] + K20 * in0 + K21 * in1;
      x[3] = xp[3] + K30 * in0 + K31 * in1;

      // ---- P_new = P_pred - K S K^T, symmetrized like the reference ----
      float T00 = K00 * S00 + K01 * S01, T01 = K00 * S01 + K01 * S11;
      float T10 = K10 * S00 + K11 * S01, T11 = K10 * S01 + K11 * S11;
      float T20 = K20 * S00 + K21 * S01, T21 = K20 * S01 + K21 * S11;
      float T30 = K30 * S00 + K31 * S01, T31 = K30 * S01 + K31 * S11;

      float p00 = M00 + q00 - (T00 * K00 + T01 * K01);
      float p01 = M01 + q01 - 0.5f * ((T00 * K10 + T01 * K11) + (T10 * K00 + T11 * K01));
      float p02 = M02 + q02 - 0.5f * ((T00 * K20 + T01 * K21) + (T20 * K00 + T21 * K01));
      float p03 = M03 + q03 - 0.5f * ((T00 * K30 + T01 * K31) + (T30 * K00 + T31 * K01));
      float p11 = M11 + q11 - (T10 * K10 + T11 * K11);
      float p12 = M12 + q12 - 0.5f * ((T10 * K20 + T11 * K21) + (T20 * K10 + T21 * K11));
      float p13 = M13 + q13 - 0.5f * ((T10 * K30 + T11 * K31) + (T30 * K10 + T31 * K11));
      float p22 = M22 + q22 - (T20 * K20 + T21 * K21);
      float p23 = M23 + q23 - 0.5f * ((T20 * K30 + T21 * K31) + (T30 * K20 + T31 * K21));
      float p33 = M33 + q33 - (T30 * K30 + T31 * K31);

      P[0][0] = p00; P[0][1] = p01; P[0][2] = p02; P[0][3] = p03;
      P[1][0] = p01; P[1][1] = p11; P[1][2] = p12; P[1][3] = p13;
      P[2][0] = p02; P[2][1] = p12; P[2][2] = p22; P[2][3] = p23;
      P[3][0] = p03; P[3][1] = p13; P[3][2] = p23; P[3][3] = p33;

      prd[0][j] = xp[0]; prd[1][j] = xp[1];  // z_pred == x_pred[:2]
      st[0][j] = x[0]; st[1][j] = x[1]; st[2][j] = x[2]; st[3][j] = x[3];
      cv[0][j] = p00; cv[1][j] = p01; cv[2][j] = p02; cv[3][j] = p03;
      cv[4][j] = p11; cv[5][j] = p12; cv[6][j] = p13;
      cv[7][j] = p22; cv[8][j] = p23; cv[9][j] = p33;
    }

    // ---- 16B-aligned non-temporal b128 stores (write-once streams) ----
    {
      float* pb = outPred + (size_t)b * 2 * (size_t)S + (size_t)c * 4;
      st_nt(pb,     mkf4(prd[0]));
      st_nt(pb + S, mkf4(prd[1]));
      float* sb = outState + (size_t)b * 4 * (size_t)S + (size_t)c * 4;
#pragma unroll
      for (int k = 0; k < 4; ++k) st_nt(sb + k * S, mkf4(st[k]));
      float* cb = outCov + (size_t)b * 16 * (size_t)S + (size_t)c * 4;
      const int symmap[16] = {0, 1, 2, 3, 1, 4, 5, 6, 2, 5, 7, 8, 3, 6, 8, 9};
#pragma unroll
      for (int e = 0; e < 16; ++e) st_nt(cb + e * S, mkf4(cv[symmap[e]]));
    }

#ifdef UKF_TDM
#ifdef UKF_TCNT
    __builtin_amdgcn_s_wait_tensorcnt(0);  // next chunk's tile is resident
#endif
    asm volatile("" ::: "memory");
#endif
  }
}

extern "C" void kernel_launch(void* const* d_in, const int* in_sizes, int n_in,
                              void* d_out, int out_size, void* d_ws, size_t ws_size,
                              hipStream_t stream) {
  (void)n_in; (void)out_size; (void)d_ws; (void)ws_size;
  const float* meas   = (const float*)d_in[0];
  const float* state0 = (const float*)d_in[1];
  const float* cov0   = (const float*)d_in[2];
  const float* ctrl   = (const float*)d_in[3];
  const float* Q      = (const float*)d_in[4];
  const float* R      = (const float*)d_in[5];

  const int B = in_sizes[1] / 4;           // state is (B,4)
  const int S = in_sizes[0] / (2 * B);     // measurements are (B,2,S)

  float* out      = (float*)d_out;
  float* outPred  = out;
  float* outState = out + (size_t)B * 2 * (size_t)S;
  float* outCov   = outState + (size_t)B * 4 * (size_t)S;

  dim3 grid((unsigned)((B + 255) / 256)), block(256);
  ukf_cdna5_kernel<<<grid, block, 0, stream>>>(meas, state0, cov0, ctrl, Q, R,
                                               outPred, outState, outCov, B, S);
}